// TransformerConvGrapHiC_35648228557521
// MI455X (gfx1250) — compile-verified
//
#include <hip/hip_runtime.h>

// Problem constants (match reference)
#define K_N   16384
#define K_E   262144
#define K_H   4
#define K_D   64
#define K_HD  256      // H*D
#define K_B   64
#define K_NPG 256
#define K_EMB 64

typedef __attribute__((ext_vector_type(2))) float v2f;
typedef __attribute__((ext_vector_type(8))) float v8f;

__device__ __forceinline__ float sigf(float x) { return 1.0f / (1.0f + expf(-x)); }

// Order-preserving float<->uint encoding for atomic max on floats.
__device__ __forceinline__ unsigned f2ord(float x) {
    unsigned u = __float_as_uint(x);
    return (u & 0x80000000u) ? ~u : (u | 0x80000000u);
}
__device__ __forceinline__ float ord2f(unsigned k) {
    unsigned u = (k & 0x80000000u) ? (k & 0x7fffffffu) : ~k;
    return __uint_as_float(u);
}

// ---------------------------------------------------------------------------
// Generic f32 GEMM via V_WMMA_F32_16X16X4_F32.
// C[M x Nc] = A[M x Kd] * B[Kd x Nc] (+ bias, optional relu, optional BN).
// Block = 256 threads = 8 waves; wave -> one 16x16 C tile; block tile 32Mx64N.
// WMMA f32 layouts per CDNA5 ISA:
//   A 16x4:  lanes 0-15 hold M=lane {K=k,k+1}; lanes 16-31 hold {K=k+2,k+3}
//   B 4x16:  lanes 0-15 hold N=lane rows {k,k+1}; lanes 16-31 rows {k+2,k+3}
//   C/D:     VGPR r -> M = r (lanes 0-15) / r+8 (lanes 16-31), N = lane&15
// mode: 0 = +bias ; 1 = relu(+bias) ; 2 = BN(relu(+bias))
// ---------------------------------------------------------------------------
__global__ __launch_bounds__(256)
void gemm_f32_wmma(const float* __restrict__ A, int lda,
                   const float* __restrict__ Bm, int ldb,
                   const float* __restrict__ bias,
                   const float* __restrict__ bng, const float* __restrict__ bnb,
                   const float* __restrict__ bnm, const float* __restrict__ bnv,
                   float* __restrict__ C, int ldc,
                   int M, int Nc, int Kd, int mode)
{
    const int lane = threadIdx.x & 31;
    const int w    = threadIdx.x >> 5;
    const int tM   = blockIdx.y * 32 + (w >> 2) * 16;
    const int tN   = blockIdx.x * 64 + (w & 3) * 16;
    if (tM >= M || tN >= Nc) return;   // wave-uniform

    const int ml   = lane & 15;
    const int hi   = lane >> 4;        // 0 or 1
    const int arow = tM + ml;
    const int bcol = tN + ml;

    v8f acc = {};
    for (int k = 0; k < Kd; k += 4) {
        const int ka = k + hi * 2;
        const float2 av = *reinterpret_cast<const float2*>(&A[arow * lda + ka]);
        v2f a; a[0] = av.x; a[1] = av.y;
        v2f b; b[0] = Bm[ka * ldb + bcol]; b[1] = Bm[(ka + 1) * ldb + bcol];
        acc = __builtin_amdgcn_wmma_f32_16x16x4_f32(
                  false, a, false, b, (short)0, acc, false, false);
    }

    const float bv = bias ? bias[bcol] : 0.0f;
    float sc = 1.0f, mb = 0.0f, bb2 = 0.0f;
    if (mode == 2) {
        sc  = rsqrtf(bnv[bcol] + 1e-5f) * bng[bcol];
        mb  = bnm[bcol];
        bb2 = bnb[bcol];
    }
#pragma unroll
    for (int r = 0; r < 8; ++r) {
        const int row = tM + hi * 8 + r;
        float y = acc[r] + bv;
        if (mode >= 1) y = fmaxf(y, 0.0f);
        if (mode == 2) y = (y - mb) * sc + bb2;
        C[row * ldc + bcol] = y;
    }
}

// ---------------------------------------------------------------------------
// Layer-0 projections: input dim = 1, so q/k/v/s are outer products.
// ---------------------------------------------------------------------------
__global__ __launch_bounds__(256)
void proj0_kernel(const float* __restrict__ x,
                  const float* __restrict__ Wq, const float* __restrict__ bq,
                  const float* __restrict__ Wk, const float* __restrict__ bk,
                  const float* __restrict__ Wv, const float* __restrict__ bv,
                  const float* __restrict__ Ws, const float* __restrict__ bs,
                  float* __restrict__ Q, float* __restrict__ Kb,
                  float* __restrict__ Vb, float* __restrict__ Sb)
{
    const int i = blockIdx.x * blockDim.x + threadIdx.x;
    if (i >= K_N * K_HD) return;
    const int n = i >> 8, j = i & 255;
    const float xv = x[n];
    Q [i] = xv * Wq[j] + bq[j];
    Kb[i] = xv * Wk[j] + bk[j];
    Vb[i] = xv * Wv[j] + bv[j];
    Sb[i] = xv * Ws[j] + bs[j];
}

// ---------------------------------------------------------------------------
// Edge logits: one wave per edge. lane covers 8 contiguous channels (one head
// per 8-lane group). logits[e,h] = dot(q[dst,h], k[src,h] + ea[e]*We[h]) / 8
// Also atomically builds per-(dst,h) max with ordered-uint keys.
// ---------------------------------------------------------------------------
__global__ __launch_bounds__(256)
void edge_logits_kernel(const int* __restrict__ ei, const float* __restrict__ ea,
                        const float* __restrict__ Q, const float* __restrict__ Kb,
                        const float* __restrict__ We,
                        float* __restrict__ LOG, unsigned* __restrict__ MAXB)
{
    const int e = blockIdx.x * 8 + (threadIdx.x >> 5);
    if (e >= K_E) return;
    const int lane = threadIdx.x & 31;
    const int src = ei[e], dst = ei[K_E + e];
    const float s = ea[e];
    const int j0 = lane * 8;
    float acc = 0.0f;
#pragma unroll
    for (int t = 0; t < 8; ++t) {
        const int j = j0 + t;
        acc += Q[dst * K_HD + j] * (Kb[src * K_HD + j] + s * We[j]);
    }
    acc += __shfl_xor(acc, 1, 32);
    acc += __shfl_xor(acc, 2, 32);
    acc += __shfl_xor(acc, 4, 32);
    if ((lane & 7) == 0) {
        const int h = lane >> 3;
        const float lg = acc * 0.125f;   // 1/sqrt(64)
        LOG[e * K_H + h] = lg;
        atomicMax(&MAXB[dst * K_H + h], f2ord(lg));
    }
}

// ex = exp(logit - max[dst,h]); accumulate segment sums.
__global__ __launch_bounds__(256)
void edge_exp_kernel(const int* __restrict__ ei, float* __restrict__ LOG,
                     const unsigned* __restrict__ MAXB, float* __restrict__ SUMB)
{
    const int i = blockIdx.x * blockDim.x + threadIdx.x;
    if (i >= K_E * K_H) return;
    const int e = i >> 2, h = i & 3;
    const int dst = ei[K_E + e];
    const float m  = ord2f(MAXB[dst * K_H + h]);
    const float ex = expf(LOG[i] - m);
    LOG[i] = ex;
    atomicAdd(&SUMB[dst * K_H + h], ex);
}

// Aggregate: OUT[dst] += a_e * (v[src] + ea*We). One wave per edge.
__global__ __launch_bounds__(256)
void edge_agg_kernel(const int* __restrict__ ei, const float* __restrict__ ea,
                     const float* __restrict__ Vb, const float* __restrict__ We,
                     const float* __restrict__ LOG, const float* __restrict__ SUMB,
                     float* __restrict__ OUT)
{
    const int e = blockIdx.x * 8 + (threadIdx.x >> 5);
    if (e >= K_E) return;
    const int lane = threadIdx.x & 31;
    const int src = ei[e], dst = ei[K_E + e];
    const float s = ea[e];
    const int h = lane >> 3;
    const float a = LOG[e * K_H + h] / (SUMB[dst * K_H + h] + 1e-16f);
    const int j0 = lane * 8;
#pragma unroll
    for (int t = 0; t < 8; ++t) {
        const int j = j0 + t;
        atomicAdd(&OUT[dst * K_HD + j], a * (Vb[src * K_HD + j] + s * We[j]));
    }
}

// beta gating: one wave per node; 768-wide dot reduced over the wave.
__global__ __launch_bounds__(256)
void beta_kernel(const float* __restrict__ OUT, const float* __restrict__ Sb,
                 const float* __restrict__ Wb, float* __restrict__ G)
{
    const int n = blockIdx.x * 8 + (threadIdx.x >> 5);
    if (n >= K_N) return;
    const int lane = threadIdx.x & 31;
    const int j0 = lane * 8;
    float ov[8], xv[8];
    float p = 0.0f;
#pragma unroll
    for (int t = 0; t < 8; ++t) {
        const int j = j0 + t;
        ov[t] = OUT[n * K_HD + j];
        xv[t] = Sb[n * K_HD + j];
        p += ov[t] * Wb[j] + xv[t] * Wb[256 + j] + (ov[t] - xv[t]) * Wb[512 + j];
    }
    p += __shfl_xor(p, 1, 32);
    p += __shfl_xor(p, 2, 32);
    p += __shfl_xor(p, 4, 32);
    p += __shfl_xor(p, 8, 32);
    p += __shfl_xor(p, 16, 32);
    const float beta = sigf(p);
#pragma unroll
    for (int t = 0; t < 8; ++t)
        G[n * K_HD + j0 + t] = beta * xv[t] + (1.0f - beta) * ov[t];
}

// ---------------------------------------------------------------------------
// Set2Set
// ---------------------------------------------------------------------------
__global__ __launch_bounds__(256)
void s2s_lstm_kernel(const float* __restrict__ Wih, const float* __restrict__ Whh,
                     const float* __restrict__ bih, const float* __restrict__ bhh,
                     float* __restrict__ QSTAR, float* __restrict__ HS,
                     float* __restrict__ CS, float* __restrict__ GATES)
{
    const int t = threadIdx.x;
    for (int idx = t; idx < K_B * 4 * K_EMB; idx += 256) {
        const int gi = idx >> 8, j = idx & 255;
        float acc = bih[j] + bhh[j];
        const float* qs = &QSTAR[gi * 2 * K_EMB];
        for (int k = 0; k < 2 * K_EMB; ++k) acc += qs[k] * Wih[k * 256 + j];
        const float* hv = &HS[gi * K_EMB];
        for (int k = 0; k < K_EMB; ++k) acc += hv[k] * Whh[k * 256 + j];
        GATES[idx] = acc;
    }
    __threadfence();
    __syncthreads();
    for (int idx = t; idx < K_B * K_EMB; idx += 256) {
        const int gi = idx >> 6, d = idx & 63;
        const float* g = &GATES[gi * 256];
        const float i_ = g[d], f_ = g[64 + d], gg = g[128 + d], o_ = g[192 + d];
        const float c = sigf(f_) * CS[idx] + sigf(i_) * tanhf(gg);
        const float h = sigf(o_) * tanhf(c);
        CS[idx] = c;
        HS[idx] = h;
        QSTAR[gi * 2 * K_EMB + d] = h;   // first half of q_star
    }
}

// e[n] = dot(x[n], h[batch[n]])  -- one wave per node (2 channels/lane)
__global__ __launch_bounds__(256)
void s2s_energy_kernel(const float* __restrict__ X, const int* __restrict__ batch,
                       const float* __restrict__ HS, float* __restrict__ EV)
{
    const int n = blockIdx.x * 8 + (threadIdx.x >> 5);
    if (n >= K_N) return;
    const int lane = threadIdx.x & 31;
    const int b = batch[n];
    const int d0 = lane * 2;
    float acc = X[n * K_EMB + d0] * HS[b * K_EMB + d0]
              + X[n * K_EMB + d0 + 1] * HS[b * K_EMB + d0 + 1];
    acc += __shfl_xor(acc, 1, 32);
    acc += __shfl_xor(acc, 2, 32);
    acc += __shfl_xor(acc, 4, 32);
    acc += __shfl_xor(acc, 8, 32);
    acc += __shfl_xor(acc, 16, 32);
    if (lane == 0) EV[n] = acc;
}

// Per-graph softmax + readout r; graphs are contiguous blocks of 256 nodes.
__global__ __launch_bounds__(256)
void s2s_readout_kernel(const float* __restrict__ EV, const float* __restrict__ X,
                        float* __restrict__ QSTAR)
{
    __shared__ float red[256];
    __shared__ float abuf[256];
    const int g = blockIdx.x, t = threadIdx.x;
    const float v = EV[g * K_NPG + t];
    red[t] = v;
    __syncthreads();
    for (int s = 128; s > 0; s >>= 1) {
        if (t < s) red[t] = fmaxf(red[t], red[t + s]);
        __syncthreads();
    }
    const float mx = red[0];
    __syncthreads();
    const float ex = expf(v - mx);
    red[t] = ex;
    __syncthreads();
    for (int s = 128; s > 0; s >>= 1) {
        if (t < s) red[t] += red[t + s];
        __syncthreads();
    }
    const float sum = red[0];
    abuf[t] = ex / (sum + 1e-16f);
    __syncthreads();
    if (t < K_EMB) {
        float r = 0.0f;
        for (int n = 0; n < K_NPG; ++n)
            r += abuf[n] * X[(g * K_NPG + n) * K_EMB + t];
        QSTAR[g * 2 * K_EMB + K_EMB + t] = r;   // second half of q_star
    }
}

// out[i,j] = sigmoid(dot(o2[i], o2[j])) -- one wave per (i,j)
__global__ __launch_bounds__(256)
void gram_kernel(const float* __restrict__ O2, float* __restrict__ out)
{
    const int w = blockIdx.x * 8 + (threadIdx.x >> 5);
    if (w >= K_B * K_B) return;
    const int lane = threadIdx.x & 31;
    const int i = w >> 6, j = w & 63;
    float acc = 0.0f;
#pragma unroll
    for (int t = 0; t < 8; ++t) {
        const int c = lane * 8 + t;
        acc += O2[i * 256 + c] * O2[j * 256 + c];
    }
    acc += __shfl_xor(acc, 1, 32);
    acc += __shfl_xor(acc, 2, 32);
    acc += __shfl_xor(acc, 4, 32);
    acc += __shfl_xor(acc, 8, 32);
    acc += __shfl_xor(acc, 16, 32);
    if (lane == 0) out[w] = sigf(acc);
}

// ---------------------------------------------------------------------------
// Host-side orchestration
// ---------------------------------------------------------------------------
extern "C" void kernel_launch(void* const* d_in, const int* in_sizes, int n_in,
                              void* d_out, int out_size, void* d_ws, size_t ws_size,
                              hipStream_t stream)
{
    (void)in_sizes; (void)n_in; (void)out_size; (void)ws_size;

    const float* x     = (const float*)d_in[0];
    const float* ea    = (const float*)d_in[1];
    const int*   ei    = (const int*)d_in[2];
    const int*   batch = (const int*)d_in[3];
    auto P = [&](int i) { return (const float*)d_in[i]; };

    // conv layer param offsets (insertion order of the reference dicts)
    // base = 4 + 16*l : Wq bq Wk bk Wv bv We Ws bs Wb Wt bt bn_g bn_b bn_m bn_v
    const int LSTM = 4 + 5 * 16;   // 84: Wih Whh bih bhh
    const int HEAD = LSTM + 4;     // 88: mu_W mu_b d0_W d0_b d1_W d1_b d2_W d2_b

    // Workspace carve-up (floats)
    float* ws = (float*)d_ws;
    size_t off = 0;
    float* HA   = ws + off; off += (size_t)K_N * K_EMB;
    float* HB   = ws + off; off += (size_t)K_N * K_EMB;
    float* Q    = ws + off; off += (size_t)K_N * K_HD;
    float* Kb   = ws + off; off += (size_t)K_N * K_HD;
    float* Vb   = ws + off; off += (size_t)K_N * K_HD;
    float* Sb   = ws + off; off += (size_t)K_N * K_HD;
    float* OUTB = ws + off; off += (size_t)K_N * K_HD;
    float* LOG  = ws + off; off += (size_t)K_E * K_H;
    unsigned* MAXB = (unsigned*)(ws + off); off += (size_t)K_N * K_H;
    float* SUMB = ws + off; off += (size_t)K_N * K_H;
    float* QSTAR= ws + off; off += (size_t)K_B * 2 * K_EMB;
    float* HS   = ws + off; off += (size_t)K_B * K_EMB;
    float* CS   = ws + off; off += (size_t)K_B * K_EMB;
    float* GATES= ws + off; off += (size_t)K_B * 4 * K_EMB;
    float* EV   = ws + off; off += (size_t)K_N;
    float* Z    = ws + off; off += (size_t)K_B * 128;
    float* O0   = ws + off; off += (size_t)K_B * 256;
    float* O1   = ws + off; off += (size_t)K_B * 512;
    float* O2   = ws + off; off += (size_t)K_B * 256;
    float* G = Q;  // beta-gated features reuse Q (dead after logits)

    auto gemm = [&](const float* A, int lda, const float* Bm, int ldb,
                    const float* bias, const float* g, const float* b2,
                    const float* m, const float* v,
                    float* C, int ldc, int M, int Nc, int Kd, int mode) {
        dim3 grid(Nc / 64, M / 32, 1);
        gemm_f32_wmma<<<grid, 256, 0, stream>>>(A, lda, Bm, ldb, bias,
                                                g, b2, m, v, C, ldc, M, Nc, Kd, mode);
    };

    float* Hbuf[2] = {HA, HB};

    for (int l = 0; l < 5; ++l) {
        const int base = 4 + 16 * l;
        const float *Wq = P(base+0), *bq = P(base+1), *Wk = P(base+2), *bk = P(base+3);
        const float *Wv = P(base+4), *bv = P(base+5), *We = P(base+6);
        const float *Wsp= P(base+7), *bs = P(base+8), *Wb = P(base+9);
        const float *Wt = P(base+10),*bt = P(base+11);
        const float *bng= P(base+12),*bnb= P(base+13),*bnm= P(base+14),*bnv= P(base+15);

        // projections q,k,v,s
        if (l == 0) {
            proj0_kernel<<<(K_N * K_HD) / 256, 256, 0, stream>>>(
                x, Wq, bq, Wk, bk, Wv, bv, Wsp, bs, Q, Kb, Vb, Sb);
        } else {
            const float* hin = Hbuf[(l + 1) & 1];
            gemm(hin, K_EMB, Wq,  K_HD, bq, 0,0,0,0, Q,  K_HD, K_N, K_HD, K_EMB, 0);
            gemm(hin, K_EMB, Wk,  K_HD, bk, 0,0,0,0, Kb, K_HD, K_N, K_HD, K_EMB, 0);
            gemm(hin, K_EMB, Wv,  K_HD, bv, 0,0,0,0, Vb, K_HD, K_N, K_HD, K_EMB, 0);
            gemm(hin, K_EMB, Wsp, K_HD, bs, 0,0,0,0, Sb, K_HD, K_N, K_HD, K_EMB, 0);
        }

        // clear atomic accumulators
        hipMemsetAsync(MAXB, 0, (size_t)K_N * K_H * 4, stream);
        hipMemsetAsync(SUMB, 0, (size_t)K_N * K_H * 4, stream);
        hipMemsetAsync(OUTB, 0, (size_t)K_N * K_HD * 4, stream);

        edge_logits_kernel<<<K_E / 8, 256, 0, stream>>>(ei, ea, Q, Kb, We, LOG, MAXB);
        edge_exp_kernel<<<(K_E * K_H) / 256, 256, 0, stream>>>(ei, LOG, MAXB, SUMB);
        edge_agg_kernel<<<K_E / 8, 256, 0, stream>>>(ei, ea, Vb, We, LOG, SUMB, OUTB);

        beta_kernel<<<K_N / 8, 256, 0, stream>>>(OUTB, Sb, Wb, G);

        // h_next = BN(relu(G @ Wt + bt))
        gemm(G, K_HD, Wt, K_EMB, bt, bng, bnb, bnm, bnv,
             Hbuf[l & 1], K_EMB, K_N, K_EMB, K_HD, 2);
    }
    float* X = Hbuf[0];   // layer 4 output lands in HA

    // ---------------- Set2Set ----------------
    hipMemsetAsync(QSTAR, 0, (size_t)K_B * 2 * K_EMB * 4, stream);
    hipMemsetAsync(HS,    0, (size_t)K_B * K_EMB * 4, stream);
    hipMemsetAsync(CS,    0, (size_t)K_B * K_EMB * 4, stream);
    for (int step = 0; step < 4; ++step) {
        s2s_lstm_kernel<<<1, 256, 0, stream>>>(P(LSTM+0), P(LSTM+1), P(LSTM+2), P(LSTM+3),
                                               QSTAR, HS, CS, GATES);
        s2s_energy_kernel<<<K_N / 8, 256, 0, stream>>>(X, batch, HS, EV);
        s2s_readout_kernel<<<K_B, 256, 0, stream>>>(EV, X, QSTAR);
    }

    // ---------------- Head MLP ----------------
    gemm(QSTAR, 128, P(HEAD+0), 128, P(HEAD+1), 0,0,0,0, Z,  128, K_B, 128, 128, 0);
    gemm(Z,     128, P(HEAD+2), 256, P(HEAD+3), 0,0,0,0, O0, 256, K_B, 256, 128, 1);
    gemm(O0,    256, P(HEAD+4), 512, P(HEAD+5), 0,0,0,0, O1, 512, K_B, 512, 256, 1);
    gemm(O1,    512, P(HEAD+6), 256, P(HEAD+7), 0,0,0,0, O2, 256, K_B, 256, 512, 1);

    gram_kernel<<<(K_B * K_B) / 8, 256, 0, stream>>>(O2, (float*)d_out);
}